// ST_mapping_44805098831884
// MI455X (gfx1250) — compile-verified
//
#include <hip/hip_runtime.h>

// ---------------------------------------------------------------------------
// ST_mapping for MI455X (gfx1250, wave32).
// 131072 independent 8x8 problems; two problems packed block-diagonally into
// each 16x16 WMMA tile; all 7 matmuls per pair via v_wmma_f32_16x16x4_f32
// (K=16 in 4 chained steps). Operands live in LDS as full 16x16 row-major
// tiles (off-diagonal blocks zeroed once), with B-operands stored transposed
// so every fetch is one unconditional ds_load_b64 at an immediate offset.
// 8x8 SPD inverse via lane-parallel Gauss-Jordan with LDS pivot broadcast
// (wave-local, in-order DS, no workgroup barriers).
// ---------------------------------------------------------------------------

typedef __attribute__((ext_vector_type(2))) float v2f;
typedef __attribute__((ext_vector_type(4))) float v4f;
typedef __attribute__((ext_vector_type(8))) float v8f;

#define WAVE            32
#define WAVES_PER_BLOCK 4
#define BLOCK_THREADS   (WAVE * WAVES_PER_BLOCK)
#define GRID_BLOCKS     2048
#define N_CH            32
#define RESO            256
#define NPAIR           65536   // (4 * 32*32 windows * 32 ch) / 2

struct __align__(16) WaveSmem {
  float A16[256];    // diag(A_p, A_q) row-major; doubles as B-op for X*A^T
  float B1[256];     // diag(B1_p, B1_q) row-major (A-operand for T1)
  float B2[256];     // diag(B2_p, B2_q) row-major (A-operand for T2)
  float B1t[256];    // transpose (B-operand for O2)
  float B2t[256];    // transpose (B-operand for O1)
  float Gt[256];     // G = A A^T, row-major (read by Gauss-Jordan)
  float TF1[256];    // T1, later overwritten by F1 (A-operands)
  float TF2[256];    // T2, later overwritten by F2
  float Ginvt[256];  // Ginv transposed (B-operand for F1/F2)
  float piv[32];     // Gauss-Jordan pivot row broadcast (g[8] | inv[8]) x2
};

__device__ __forceinline__ void ldsfence() {
#if defined(__AMDGCN__)
  __builtin_amdgcn_wave_barrier();
#endif
}

// One 16x16x16 matmul as 4 chained wmma_f32_16x16x4.
// A-operand: lanes 0-15 = rows M with K pair {0,1}; lanes 16-31 = K pair {2,3}.
// Both tiles are read with the same contiguous pattern (Bt holds W^T).
// lb = l15*16 + half*2  (precomputed per lane).
__device__ __forceinline__ v8f mm16(const float* At, const float* Bt, int lb) {
  v8f acc;
#pragma unroll
  for (int i = 0; i < 8; ++i) acc[i] = 0.0f;
#if defined(__AMDGCN__)
#pragma unroll
  for (int step = 0; step < 4; ++step) {
    v2f a = *(const v2f*)(At + lb + step * 4);
    v2f b = *(const v2f*)(Bt + lb + step * 4);
    acc = __builtin_amdgcn_wmma_f32_16x16x4_f32(false, a, false, b,
                                                (short)0, acc, false, false);
  }
#endif
  return acc;
}

// Full unconditional D-layout -> row-major 16x16 tile store (off-diag is
// exactly zero for block-diagonal products, so this preserves zero structure).
__device__ __forceinline__ void storeTile(const v8f& d, float* tile,
                                          int half, int l15) {
  const int m0 = half * 8;
#pragma unroll
  for (int v = 0; v < 8; ++v) tile[(m0 + v) * 16 + l15] = d[v];
}

// Scatter the two diagonal 8x8 result blocks straight to global memory.
__device__ __forceinline__ void storeOut(const v8f& d, float* __restrict__ out,
                                         size_t basep, size_t baseq,
                                         int half, int l15) {
  if (half == 0) {
    if (l15 < 8) {
#pragma unroll
      for (int v = 0; v < 8; ++v) out[basep + (size_t)v * RESO + l15] = d[v];
    }
  } else {
    if (l15 >= 8) {
#pragma unroll
      for (int v = 0; v < 8; ++v) out[baseq + (size_t)v * RESO + (l15 - 8)] = d[v];
    }
  }
}

// Output base for problem e in channel block chBase (0=time, 32=spatial).
// Faithful to the reference's view/permute: out[b, chBase+hy, wx*8+r, c*8+s].
__device__ __forceinline__ size_t outBase(int e, int chBase) {
  const int c  = e & 31;
  const int n  = e >> 5;
  const int b  = n >> 10;
  const int hy = (n >> 5) & 31;
  const int wx = n & 31;
  return (((size_t)(b * 64 + chBase + hy) * RESO) + (size_t)(wx * 8)) * RESO
         + (size_t)(c * 8);
}

// Pointer to one 8-float window row for flat row id rf (mat in [0,6), r in [0,8)).
__device__ __forceinline__ const float* rowPtr(const float* __restrict__ c0,
                                               const float* __restrict__ c1,
                                               const float* __restrict__ f0,
                                               int ep, int eq, int rf) {
  const int mat = rf >> 3;
  const int r   = rf & 7;
  const float* src = (mat < 2) ? c0 : ((mat < 4) ? c1 : f0);
  const int e  = (mat & 1) ? eq : ep;
  const int c  = e & 31;
  const int n  = e >> 5;
  const int b  = n >> 10;
  const int hy = (n >> 5) & 31;
  const int wx = n & 31;
  return src + (((size_t)(b * N_CH + c) * RESO + (size_t)(hy * 8 + r)) * RESO
                + (size_t)(wx * 8));
}

__global__ __launch_bounds__(BLOCK_THREADS)
void st_mapping_kernel(const float* __restrict__ c0,
                       const float* __restrict__ c1,
                       const float* __restrict__ f0,
                       float* __restrict__ out) {
  __shared__ WaveSmem smem[WAVES_PER_BLOCK];
  const int lane = threadIdx.x & (WAVE - 1);
  const int wid  = threadIdx.x >> 5;
  WaveSmem* S = &smem[wid];
  const int half = lane >> 4;
  const int l15  = lane & 15;
  const int lb   = l15 * 16 + half * 2;   // per-lane operand fetch base
  const int waveGlobal = blockIdx.x * WAVES_PER_BLOCK + wid;
  const int waveStride = GRID_BLOCKS * WAVES_PER_BLOCK;

  // One-time zero of the whole wave-private LDS region (keeps the never-
  // written off-diagonal 8x8 blocks of all tiles at exactly zero).
  {
    v4f z;
#pragma unroll
    for (int i = 0; i < 4; ++i) z[i] = 0.0f;
    v4f* base = (v4f*)S;
    const int nvec = (int)(sizeof(WaveSmem) / sizeof(v4f));
    for (int i = lane; i < nvec; i += WAVE) base[i] = z;
  }
  ldsfence();

  for (int pair = waveGlobal; pair < NPAIR; pair += waveStride) {
    const int ep = pair * 2;
    const int eq = ep + 1;

    // ---- Stage the 6 window matrices (48 rows of 8 floats) into LDS ----
#pragma unroll
    for (int it = 0; it < 2; ++it) {
      const int rf = it * 32 + lane;
      if (rf < 48) {
        const int mat = rf >> 3;
        const int r   = rf & 7;
        const int q   = mat & 1;
        const float* rowp = rowPtr(c0, c1, f0, ep, eq, rf);
        v4f lo = *(const v4f*)(rowp);
        v4f hi = *(const v4f*)(rowp + 4);
        float* tile = (mat < 2) ? S->A16 : ((mat < 4) ? S->B1 : S->B2);
        const int rb = q ? ((8 + r) * 16 + 8) : (r * 16);
        *(v4f*)(tile + rb)     = lo;
        *(v4f*)(tile + rb + 4) = hi;
        if (mat >= 2) {                 // transposed copies for B-operand use
          float* tt = (mat < 4) ? S->B1t : S->B2t;
          float tmp[8];
          *(v4f*)(tmp)     = lo;
          *(v4f*)(tmp + 4) = hi;
          const int co = q ? 8 : 0;
#pragma unroll
          for (int s = 0; s < 8; ++s) tt[(co + s) * 16 + co + r] = tmp[s];
        }
      }
    }
    // Prefetch next pair's rows into cache while we compute.
    {
      const int nxt = pair + waveStride;
      if (nxt < NPAIR && lane < 48) {
        __builtin_prefetch(rowPtr(c0, c1, f0, nxt * 2, nxt * 2 + 1, lane), 0, 0);
      }
    }
    ldsfence();

    // G = A A^T ; T1 = C1 A^T ; T2 = F0 A^T  (B-op of X*A^T is A^T, whose
    // transpose is A itself -> reuse the row-major A16 tile directly).
    v8f G  = mm16(S->A16, S->A16, lb);
    v8f T1 = mm16(S->B1,  S->A16, lb);
    v8f T2 = mm16(S->B2,  S->A16, lb);
    storeTile(G,  S->Gt,  half, l15);
    storeTile(T1, S->TF1, half, l15);
    storeTile(T2, S->TF2, half, l15);
    ldsfence();

    // ---- Gauss-Jordan inverse of the two 8x8 SPD Grams ----
    // lanes (half*16 + 0..7) own one row each; pivot row broadcast via LDS.
    {
      const bool act = (l15 < 8);
      const int  co  = half ? 8 : 0;
      float gr[8], ir[8];
#pragma unroll
      for (int j = 0; j < 8; ++j) { gr[j] = 0.0f; ir[j] = 0.0f; }
      if (act) {
        const float* Gm = S->Gt + (co + l15) * 16 + co;
        *(v4f*)(gr)     = *(const v4f*)(Gm);
        *(v4f*)(gr + 4) = *(const v4f*)(Gm + 4);
        ir[l15] = 1.0f;
      }
      float* pv = S->piv + half * 16;
#pragma unroll
      for (int k = 0; k < 8; ++k) {
        if (act && l15 == k) {
          *(v4f*)(pv)      = *(v4f*)(gr);
          *(v4f*)(pv + 4)  = *(v4f*)(gr + 4);
          *(v4f*)(pv + 8)  = *(v4f*)(ir);
          *(v4f*)(pv + 12) = *(v4f*)(ir + 4);
        }
        ldsfence();
        float pg[8], pi[8];
        *(v4f*)(pg)     = *(const v4f*)(pv);
        *(v4f*)(pg + 4) = *(const v4f*)(pv + 4);
        *(v4f*)(pi)     = *(const v4f*)(pv + 8);
        *(v4f*)(pi + 4) = *(const v4f*)(pv + 12);
        const float d    = pg[k];
        const float dinv = (__builtin_fabsf(d) > 1e-20f) ? (1.0f / d) : 0.0f;
        if (act) {
          if (l15 == k) {
#pragma unroll
            for (int j = 0; j < 8; ++j) { gr[j] *= dinv; ir[j] *= dinv; }
          } else {
            const float f = gr[k] * dinv;
#pragma unroll
            for (int j = 0; j < 8; ++j) {
              gr[j] = fmaf(-f, pg[j], gr[j]);
              ir[j] = fmaf(-f, pi[j], ir[j]);
            }
          }
        }
        ldsfence();
      }
      // Write Ginv transposed (column scatter) -> ready-to-fetch B-operand.
      if (act) {
#pragma unroll
        for (int j = 0; j < 8; ++j) S->Ginvt[(co + j) * 16 + co + l15] = ir[j];
      }
    }
    ldsfence();

    // F1 = T1 Ginv ; F2 = T2 Ginv
    v8f F1 = mm16(S->TF1, S->Ginvt, lb);
    v8f F2 = mm16(S->TF2, S->Ginvt, lb);
    storeTile(F1, S->TF1, half, l15);   // safe: reads precede stores in order
    storeTile(F2, S->TF2, half, l15);
    ldsfence();

    // O1 = F1 F0mat (f1_time) ; O2 = F2 C1mat (f1_spatial)
    v8f O1 = mm16(S->TF1, S->B2t, lb);
    v8f O2 = mm16(S->TF2, S->B1t, lb);

    const size_t bt_p = outBase(ep, 0),  bt_q = outBase(eq, 0);
    const size_t bs_p = outBase(ep, 32), bs_q = outBase(eq, 32);
    storeOut(O1, out, bt_p, bt_q, half, l15);
    storeOut(O2, out, bs_p, bs_q, half, l15);
  }
}

extern "C" void kernel_launch(void* const* d_in, const int* in_sizes, int n_in,
                              void* d_out, int out_size, void* d_ws, size_t ws_size,
                              hipStream_t stream) {
  (void)in_sizes; (void)n_in; (void)out_size; (void)d_ws; (void)ws_size;
  const float* c0 = (const float*)d_in[0];
  const float* c1 = (const float*)d_in[1];
  const float* f0 = (const float*)d_in[2];
  float* out = (float*)d_out;
  st_mapping_kernel<<<dim3(GRID_BLOCKS), dim3(BLOCK_THREADS), 0, stream>>>(
      c0, c1, f0, out);
}